// residualBlock_nomask_72919954752306
// MI455X (gfx1250) — compile-verified
//
#include <hip/hip_runtime.h>
#include <hip/hip_bf16.h>

// CDNA5 (gfx1250) sparse residual block.
//   u   = elu(elu(feats) @ Wa)          [N,64]   (stored A-fragment-permuted)
//   c   = sum_k gather(u, nb_k) @ Wb[k] [N,64]   (WMMA f32 16x16x4, sparsity-skipped)
//   out = elu(c) @ Wc + feats           [N,128]
//
// H = 64, 2H = 128, KVOL = 125.

typedef float v2f __attribute__((ext_vector_type(2)));
typedef float v8f __attribute__((ext_vector_type(8)));

#define H      64
#define H2     128
#define KVOL   125

__device__ __forceinline__ float elu1(float x) {
    return x > 0.0f ? x : expm1f(x);
}

// A-row permutation: original channel K -> packed position p so that
// lane<16 reads floats [0..31] (K%4 in {0,1}) and lane>=16 reads [32..63]
// (K%4 in {2,3}), each contiguous. Per WMMA K-chunk c, lane slice is
// pair (2c, 2c+1) of its half.
__device__ __forceinline__ int a_perm(int k) {
    return ((k & 2) ? 32 : 0) + ((k >> 2) * 2) + (k & 1);
}

// ---------------------------------------------------------------------------
// K0: permute Wb [125][64][64] into B-fragment order.
// wbp[((k*4 + ct)*16 + c)*32 + lane] = { B[r0][n], B[r0+1][n] }
//   lane<16 : r0 = 4c,   n = ct*16 + lane
//   lane>=16: r0 = 4c+2, n = ct*16 + lane-16
// ---------------------------------------------------------------------------
__global__ void k0_permute_wb(const float* __restrict__ Wb,
                              float2* __restrict__ wbp) {
    int tid = blockIdx.x * blockDim.x + threadIdx.x;         // 125*4*16*32
    if (tid >= KVOL * 4 * 16 * 32) return;
    int lane = tid & 31;
    int c    = (tid >> 5) & 15;
    int ct   = (tid >> 9) & 3;
    int k    = tid >> 11;
    int r0   = 4 * c + ((lane < 16) ? 0 : 2);
    int n    = ct * 16 + (lane & 15);
    const float* Wk = Wb + (size_t)k * H * H;
    wbp[tid] = make_float2(Wk[(size_t)r0 * H + n], Wk[(size_t)(r0 + 1) * H + n]);
}

// ---------------------------------------------------------------------------
// K1: u_perm = elu( elu(feats) @ Wa ), 4 rows per block iteration.
// ---------------------------------------------------------------------------
__global__ void k1_in_proj(const float* __restrict__ feats,
                           const float* __restrict__ Wa,   // [128][64]
                           float* __restrict__ u,          // [N][64] permuted
                           int N) {
    __shared__ float sWa[H2 * H];     // 32 KB
    __shared__ float sIn[4 * H2];     // 2 KB
    int tid = threadIdx.x;            // 256 threads

    #pragma unroll
    for (int j = 0; j < (H2 * H) / 256; ++j)
        sWa[j * 256 + tid] = Wa[j * 256 + tid];
    __syncthreads();

    int nq = (N + 3) / 4;
    for (int q = blockIdx.x; q < nq; q += gridDim.x) {
        int r0 = q * 4;
        // stage 4 rows of elu(feats) into LDS
        #pragma unroll
        for (int j = 0; j < 2; ++j) {
            int e   = j * 256 + tid;          // 0..511
            int row = e >> 7, col = e & 127;
            float v = (r0 + row < N) ? feats[(size_t)(r0 + row) * H2 + col] : 0.0f;
            sIn[e] = elu1(v);
        }
        __syncthreads();

        int row = tid >> 6;                   // 0..3
        int col = tid & 63;                   // 0..63
        float acc = 0.0f;
        #pragma unroll
        for (int kk = 0; kk < H2; ++kk)
            acc = fmaf(sIn[row * H2 + kk], sWa[kk * H + col], acc);
        if (r0 + row < N)
            u[(size_t)(r0 + row) * H + a_perm(col)] = elu1(acc);
        __syncthreads();
    }
}

// ---------------------------------------------------------------------------
// K2: sparse conv via WMMA f32 16x16x4. One wave per 16-point tile,
// 8 waves per block. Ballot-skip offsets with no valid neighbor in the
// tile (wave-uniform -> EXEC stays all-1s around v_wmma).
// ---------------------------------------------------------------------------
__global__ void
__launch_bounds__(256)
k2_sparse_conv(const float* __restrict__ u,        // [N][64] A-permuted
               const float2* __restrict__ wbp,     // B fragments
               const int* __restrict__ nb,         // [125][N]
               float* __restrict__ cbuf,           // [N][64]
               int N, int NT) {
    int wave = threadIdx.x >> 5;
    int lane = threadIdx.x & 31;
    int tile = blockIdx.x * 8 + wave;
    if (tile >= NT) return;

    int i0 = tile * 16;
    int r  = lane & 15;                 // tile row this lane gathers
    int gi = i0 + r;
    int half = (lane < 16) ? 0 : 32;    // which half of the permuted row

    v8f acc0 = {}, acc1 = {}, acc2 = {}, acc3 = {};

    for (int k = 0; k < KVOL; ++k) {
        int idx = (gi < N) ? nb[(size_t)k * N + gi] : -1;
        unsigned long long any = __ballot(idx >= 0);
        if (any == 0ull) continue;      // wave-uniform skip

        // gather this lane's 32-float A slice (contiguous, 8 x b128)
        float4 a4[8];
        if (idx >= 0) {
            const float4* rp =
                (const float4*)(u + (size_t)idx * H + half);
            #pragma unroll
            for (int qq = 0; qq < 8; ++qq) a4[qq] = rp[qq];
        } else {
            #pragma unroll
            for (int qq = 0; qq < 8; ++qq)
                a4[qq] = make_float4(0.f, 0.f, 0.f, 0.f);
        }

        const float2* bk = wbp + (size_t)k * 4 * 16 * 32 + lane;
        #pragma unroll
        for (int c = 0; c < 16; ++c) {
            float f0 = (c & 1) ? a4[c >> 1].z : a4[c >> 1].x;
            float f1 = (c & 1) ? a4[c >> 1].w : a4[c >> 1].y;
            v2f af; af[0] = f0; af[1] = f1;
            // 4 column tiles share this A fragment
            {
                float2 b = bk[(0 * 16 + c) * 32];
                v2f bf; bf[0] = b.x; bf[1] = b.y;
                acc0 = __builtin_amdgcn_wmma_f32_16x16x4_f32(
                    false, af, false, bf, (short)0, acc0, false, false);
            }
            {
                float2 b = bk[(1 * 16 + c) * 32];
                v2f bf; bf[0] = b.x; bf[1] = b.y;
                acc1 = __builtin_amdgcn_wmma_f32_16x16x4_f32(
                    false, af, false, bf, (short)0, acc1, false, false);
            }
            {
                float2 b = bk[(2 * 16 + c) * 32];
                v2f bf; bf[0] = b.x; bf[1] = b.y;
                acc2 = __builtin_amdgcn_wmma_f32_16x16x4_f32(
                    false, af, false, bf, (short)0, acc2, false, false);
            }
            {
                float2 b = bk[(3 * 16 + c) * 32];
                v2f bf; bf[0] = b.x; bf[1] = b.y;
                acc3 = __builtin_amdgcn_wmma_f32_16x16x4_f32(
                    false, af, false, bf, (short)0, acc3, false, false);
            }
        }
    }

    // C layout (16x16 f32): VGPR j -> lanes 0-15: (M=j, n=lane),
    // lanes 16-31: (M=j+8, n=lane-16)
    int mbase = (lane < 16) ? 0 : 8;
    int ncol  = lane & 15;
    #pragma unroll
    for (int j = 0; j < 8; ++j) {
        int row = i0 + mbase + j;
        if (row < N) {
            float* cr = cbuf + (size_t)row * H;
            cr[0 * 16 + ncol] = acc0[j];
            cr[1 * 16 + ncol] = acc1[j];
            cr[2 * 16 + ncol] = acc2[j];
            cr[3 * 16 + ncol] = acc3[j];
        }
    }
}

// ---------------------------------------------------------------------------
// K3: out = elu(c) @ Wc + feats   (2 rows per block iteration)
// ---------------------------------------------------------------------------
__global__ void k3_out_proj(const float* __restrict__ cbuf,  // [N][64]
                            const float* __restrict__ Wc,    // [64][128]
                            const float* __restrict__ feats, // [N][128]
                            float* __restrict__ out,         // [N][128]
                            int N) {
    __shared__ float sWc[H * H2];   // 32 KB
    __shared__ float sC[2 * H];     // 512 B
    int tid = threadIdx.x;          // 256 threads

    #pragma unroll
    for (int j = 0; j < (H * H2) / 256; ++j)
        sWc[j * 256 + tid] = Wc[j * 256 + tid];
    __syncthreads();

    int np = (N + 1) / 2;
    for (int p = blockIdx.x; p < np; p += gridDim.x) {
        int r0 = p * 2;
        if (tid < 2 * H) {
            int row = tid >> 6, col = tid & 63;
            float v = (r0 + row < N) ? cbuf[(size_t)(r0 + row) * H + col] : 0.0f;
            sC[tid] = elu1(v);
        }
        __syncthreads();

        int row = tid >> 7;         // 0..1
        int col = tid & 127;        // 0..127
        float acc = 0.0f;
        #pragma unroll
        for (int kk = 0; kk < H; ++kk)
            acc = fmaf(sC[row * H + kk], sWc[kk * H2 + col], acc);
        if (r0 + row < N) {
            size_t o = (size_t)(r0 + row) * H2 + col;
            out[o] = acc + feats[o];
        }
        __syncthreads();
    }
}

// ---------------------------------------------------------------------------
extern "C" void kernel_launch(void* const* d_in, const int* in_sizes, int n_in,
                              void* d_out, int out_size, void* d_ws, size_t ws_size,
                              hipStream_t stream) {
    const float* feats = (const float*)d_in[0];   // [N,128]
    const float* Wa    = (const float*)d_in[1];   // [128,64]
    const float* Wb    = (const float*)d_in[2];   // [125,64,64]
    const float* Wc    = (const float*)d_in[3];   // [64,128]
    const int*   nb    = (const int*)d_in[4];     // [125,N]
    float*       out   = (float*)d_out;

    const int N  = in_sizes[0] / H2;
    const int NT = (N + 15) / 16;

    // workspace: u_perm [N*64] | c [N*64] | wbp [125*64*64/2 float2]
    float*  u    = (float*)d_ws;
    float*  cbuf = u + (size_t)N * H;
    float2* wbp  = (float2*)(cbuf + (size_t)N * H);

    // K0: weight permute (125*4*16*32 = 256000 float2)
    k0_permute_wb<<<(KVOL * 4 * 16 * 32 + 255) / 256, 256, 0, stream>>>(Wb, wbp);

    // K1: input projection + elu, permuted store
    k1_in_proj<<<1280, 256, 0, stream>>>(feats, Wa, u, N);

    // K2: WMMA sparse conv, 8 tiles (waves) per block
    k2_sparse_conv<<<(NT + 7) / 8, 256, 0, stream>>>(u, wbp, nb, cbuf, N, NT);

    // K3: output projection + residual
    k3_out_proj<<<2048, 256, 0, stream>>>(cbuf, Wc, feats, out, N);
}